// KinematicBicycle_82214263980490
// MI455X (gfx1250) — compile-verified
//
#include <hip/hip_runtime.h>

// Kinematic bicycle rollout, H=8192, reformulated as parallel scans.
// Single workgroup (1024 thr = 32 waves on one WGP), data resident in LDS.
// f32 prefix sums run on the matrix pipe via V_WMMA_F32_16X16X4_F32:
//   tile(256) as 16x16 col-major-in-time X;  Y = L*X (4 wmma, K=4 slices)
//   then Y' = Trow(totals)*Mask + Y (4 more wmma) = full intra-tile scan.

#define HSTEPS 8192
#define BLOCK  1024
#define NWAVE  32

#define ACC_LIMIT   3.0f
#define STEER_LIMIT 0.6f
#define V_MIN_C     0.0f
#define V_MAX_C     30.0f
#define WHEELBASE   2.7f

typedef __attribute__((ext_vector_type(2))) float v2f;
typedef __attribute__((ext_vector_type(8))) float v8f;

__device__ __forceinline__ float clipf(float x, float lo, float hi) {
    return fminf(fmaxf(x, lo), hi);
}

struct SMem {
    float stage[NWAVE][256];  // 32 KB: per-wave tile staging for WMMA B operand
    float tot[NWAVE][16];     //  2 KB: per-wave column totals
    float tileSum[NWAVE];
    float tileOff[NWAVE];
    float fS[2][BLOCK];       // 24 KB: ping-pong clamp-add function scan
    float fL[2][BLOCK];
    float fH[2][BLOCK];
};

// Inclusive prefix sum of the 8192-element sequence distributed as
// v[0..7] = thread's 8 consecutive timesteps. Intra-tile (256/wave) via WMMA,
// cross-tile via a 32-entry LDS scan. aL/bM: hoisted triangular constants.
__device__ __forceinline__ void tile_scan(float v[8], SMem* sm, int wave, int lane,
                                          const v2f aL[4], const v2f bM[4]) {
    const int col  = lane & 15;   // N / segment index
    const int half = lane >> 4;   // lane half selects K/M sub-rows

    __syncthreads();
    float* st = sm->stage[wave];
#pragma unroll
    for (int r = 0; r < 8; ++r)
        st[col * 16 + half * 8 + r] = v[r];   // X[row][col], row = half*8+r
    __syncthreads();

    // Y = L * X : 4 chained wmma over K-slices of 4
    v8f c = {};
#pragma unroll
    for (int kb = 0; kb < 4; ++kb) {
        v2f b;
#pragma unroll
        for (int j = 0; j < 2; ++j)
            b[j] = st[col * 16 + kb * 4 + half * 2 + j];  // X[k][n], n = col
        c = __builtin_amdgcn_wmma_f32_16x16x4_f32(false, aL[kb], false, b,
                                                  (short)0, c, false, false);
    }

    // column totals Y[15][j] live in lanes 16..31, c[7]
    if (half == 1) sm->tot[wave][col] = c[7];
    __syncthreads();

    // Y' = A2 * B2 + Y, A2[m][k] = total[k], B2[k][j] = (k < j)
    // => adds exclusive cross-segment offset to every column.
#pragma unroll
    for (int kb = 0; kb < 4; ++kb) {
        v2f a;
#pragma unroll
        for (int j = 0; j < 2; ++j)
            a[j] = sm->tot[wave][kb * 4 + half * 2 + j];
        c = __builtin_amdgcn_wmma_f32_16x16x4_f32(false, a, false, bM[kb],
                                                  (short)0, c, false, false);
    }

    // tile total = Y'[15][15] (lane 31, c[7])
    if (lane == 31) sm->tileSum[wave] = c[7];
    __syncthreads();

    // exclusive scan of 32 tile sums (wave 0)
    if (threadIdx.x < NWAVE) {
        float tv = sm->tileSum[threadIdx.x];
        float incl = tv;
#pragma unroll
        for (int d = 1; d < 32; d <<= 1) {
            float o = __shfl_up(incl, d, 32);
            if ((int)threadIdx.x >= d) incl += o;
        }
        sm->tileOff[threadIdx.x] = incl - tv;
    }
    __syncthreads();

    const float carry = sm->tileOff[wave];
#pragma unroll
    for (int r = 0; r < 8; ++r) v[r] = c[r] + carry;  // c[r] <-> thread's r-th step
}

__global__ __launch_bounds__(BLOCK) void
KinematicBicycle_82214263980490_kernel(const float* __restrict__ x0,
                                       const float* __restrict__ U,
                                       const float* __restrict__ dtp,
                                       float* __restrict__ out) {
    __shared__ SMem sm;
    const int tid  = threadIdx.x;
    const int wave = tid >> 5;
    const int lane = tid & 31;
    const int col  = lane & 15;
    const int half = lane >> 4;
    // rank: position of this thread's 8-step chunk in time order
    const int rank = wave * 32 + col * 2 + half;
    // first timestep owned by this thread (matches WMMA D layout)
    const int t0 = wave * 256 + col * 16 + half * 8;

    // hoisted triangular WMMA constants (invariant across the three scans)
    v2f aL[4], bM[4];
#pragma unroll
    for (int kb = 0; kb < 4; ++kb) {
#pragma unroll
        for (int j = 0; j < 2; ++j) {
            const int k = kb * 4 + half * 2 + j;   // absolute K row
            aL[kb][j] = (k <= col) ? 1.0f : 0.0f;  // L[m][k],  m = col
            bM[kb][j] = (k <  col) ? 1.0f : 0.0f;  // B2[k][j], j = col
        }
    }

    const float dt = dtp[0];
    const float px = x0[0], py = x0[1], pth = x0[2], pv = x0[3];

    // ---- load controls for our 8 steps (4x float4, 64B aligned) ----
    float s[8], dl[8];
    const float4* u4 = (const float4*)(U + (size_t)t0 * 2);
#pragma unroll
    for (int q = 0; q < 4; ++q) {
        const float4 u = u4[q];
        s[2 * q]      = clipf(u.x, -ACC_LIMIT, ACC_LIMIT) * dt;
        dl[2 * q]     = clipf(u.y, -STEER_LIMIT, STEER_LIMIT);
        s[2 * q + 1]  = clipf(u.z, -ACC_LIMIT, ACC_LIMIT) * dt;
        dl[2 * q + 1] = clipf(u.w, -STEER_LIMIT, STEER_LIMIT);
    }

    // ---- phase 1: velocity = scan over clamp-add functions ----
    // per-thread sequential composition of 8 functions f_i(v)=clip(v+s_i,0,30)
    float S = s[0], Lo = V_MIN_C, Hi = V_MAX_C;
#pragma unroll
    for (int r = 1; r < 8; ++r) {
        S += s[r];
        Lo = clipf(Lo + s[r], V_MIN_C, V_MAX_C);
        Hi = clipf(Hi + s[r], V_MIN_C, V_MAX_C);
    }
    int p = 0;
    sm.fS[0][rank] = S; sm.fL[0][rank] = Lo; sm.fH[0][rank] = Hi;
    __syncthreads();
    // Hillis-Steele over 1024 composed functions, ping-pong (1 barrier/round)
    for (int off = 1; off < BLOCK; off <<= 1) {
        if (rank >= off) {   // new = mine ∘ left  (left applied first)
            const float pS = sm.fS[p][rank - off];
            const float pL = sm.fL[p][rank - off];
            const float pH = sm.fH[p][rank - off];
            const float nS = pS + S;
            const float nL = clipf(pL + S, Lo, Hi);
            const float nH = clipf(pH + S, Lo, Hi);
            S = nS; Lo = nL; Hi = nH;
        }
        p ^= 1;
        sm.fS[p][rank] = S; sm.fL[p][rank] = Lo; sm.fH[p][rank] = Hi;
        __syncthreads();
    }
    // carry-in velocity for this chunk: apply exclusive prefix function to w0
    float w = clipf(pv, V_MIN_C, V_MAX_C);
    if (rank > 0)
        w = clipf(w + sm.fS[p][rank - 1], sm.fL[p][rank - 1], sm.fH[p][rank - 1]);

    // per-step velocities + theta increments
    float wst[8], vout[8], dth[8];
    const float dtOverL = dt / WHEELBASE;
#pragma unroll
    for (int r = 0; r < 8; ++r) {
        wst[r] = w;                               // v used inside step t0+r
        dth[r] = w * tanf(dl[r]) * dtOverL;
        w = clipf(w + s[r], V_MIN_C, V_MAX_C);    // stored v for row t0+r+1
        vout[r] = w;
    }

    // ---- phase 2: theta prefix sum (WMMA scan) ----
    float incl[8];
#pragma unroll
    for (int r = 0; r < 8; ++r) incl[r] = dth[r];
    tile_scan(incl, &sm, wave, lane, aL, bM);
    float thout[8], thst[8];
#pragma unroll
    for (int r = 0; r < 8; ++r) {
        thout[r] = pth + incl[r];       // theta after step t0+r
        thst[r]  = thout[r] - dth[r];   // theta at start of step t0+r
    }

    // ---- phase 3: x and y prefix sums (WMMA scans) ----
    float iy[8];
#pragma unroll
    for (int r = 0; r < 8; ++r) {
        float sn, cs;
        sincosf(thst[r], &sn, &cs);     // shared range reduction
        const float g = wst[r] * dt;
        incl[r] = g * cs;
        iy[r]   = g * sn;
    }
    tile_scan(incl, &sm, wave, lane, aL, bM);   // x increments -> inclusive
    tile_scan(iy,   &sm, wave, lane, aL, bM);   // y increments -> inclusive

    // ---- output rows t0+1 .. t0+8 ----
#pragma unroll
    for (int r = 0; r < 8; ++r) {
        float4 row = make_float4(px + incl[r], py + iy[r], thout[r], vout[r]);
        *(float4*)(out + (size_t)(t0 + r + 1) * 4) = row;
    }
    if (tid == 0) *(float4*)out = make_float4(px, py, pth, pv);
}

extern "C" void kernel_launch(void* const* d_in, const int* in_sizes, int n_in,
                              void* d_out, int out_size, void* d_ws, size_t ws_size,
                              hipStream_t stream) {
    (void)in_sizes; (void)n_in; (void)out_size; (void)d_ws; (void)ws_size;
    const float* x0 = (const float*)d_in[0];   // [4]
    const float* U  = (const float*)d_in[1];   // [8192, 2]
    const float* dt = (const float*)d_in[2];   // scalar
    float* out = (float*)d_out;                // [8193, 4]
    KinematicBicycle_82214263980490_kernel<<<1, BLOCK, 0, stream>>>(x0, U, dt, out);
}